// SignedGCNLike_26603027432194
// MI455X (gfx1250) — compile-verified
//
#include <hip/hip_runtime.h>

#define N_NODES 100000
#define D_IN    128
#define H_DIM   128
#define E_EDGES 1600000
#define L_LAYERS 2

typedef float v2f __attribute__((ext_vector_type(2)));
typedef float v8f __attribute__((ext_vector_type(8)));

// ---------------------------------------------------------------------------
// GEMM + bias + tanh:  out[i][j] = tanh(sum_d A[i][d] * W[j][d] + b[j])
// A: [n, 128] row-major, W: [128, 128] row-major (we multiply by W^T).
// One wave per 16x16 output tile, V_WMMA_F32_16X16X4_F32, K stepped by 4.
//
// f32 WMMA fragment layouts (wave32):
//   A (16x4): lanes 0-15 -> rows M=0..15 with {K=k0,k0+1}; lanes 16-31 -> {k0+2,k0+3}
//   B (4x16): mirrored, N = lane%16
//   C/D (16x16): VGPR v -> row v + 8*(lane>=16), col = lane%16
// ---------------------------------------------------------------------------
__global__ void __launch_bounds__(256)
gemm_tanh_kernel(const float* __restrict__ A,
                 const float* __restrict__ W,
                 const float* __restrict__ b,
                 float* __restrict__ out, int n) {
  const int wave = threadIdx.x >> 5;   // 0..7 -> column tile
  const int lane = threadIdx.x & 31;
  const int half = lane >> 4;          // 0 or 1
  const int lm   = lane & 15;
  const int i0   = blockIdx.x * 16;    // row tile
  const int j0   = wave * 16;          // col tile
  if (i0 >= n) return;

  const float bias = b[j0 + lm];
  v8f acc;
#pragma unroll
  for (int v = 0; v < 8; ++v) acc[v] = bias;   // C pre-loaded with column bias

  const float* arow = A + (size_t)(i0 + lm) * D_IN + 2 * half;
  const float* wrow = W + (size_t)(j0 + lm) * D_IN + 2 * half;

#pragma unroll 8
  for (int k0 = 0; k0 < D_IN; k0 += 4) {
    v2f af = *(const v2f*)(arow + k0);
    v2f bf = *(const v2f*)(wrow + k0);
    acc = __builtin_amdgcn_wmma_f32_16x16x4_f32(
        /*neg_a=*/false, af, /*neg_b=*/false, bf,
        /*c_mod=*/(short)0, acc, /*reuse_a=*/false, /*reuse_b=*/false);
  }

#pragma unroll
  for (int v = 0; v < 8; ++v)
    out[(size_t)(i0 + v + 8 * half) * H_DIM + j0 + lm] = tanhf(acc[v]);
}

// ---------------------------------------------------------------------------
// Fused layer GEMM:  out = tanh(Ap Wp^T + bp + An Wn^T + bn)
// Two WMMA chains per K-step into a single accumulator.
// ---------------------------------------------------------------------------
__global__ void __launch_bounds__(256)
gemm2_tanh_kernel(const float* __restrict__ Ap,
                  const float* __restrict__ Wp,
                  const float* __restrict__ bp,
                  const float* __restrict__ An,
                  const float* __restrict__ Wn,
                  const float* __restrict__ bn,
                  float* __restrict__ out, int n) {
  const int wave = threadIdx.x >> 5;
  const int lane = threadIdx.x & 31;
  const int half = lane >> 4;
  const int lm   = lane & 15;
  const int i0   = blockIdx.x * 16;
  const int j0   = wave * 16;
  if (i0 >= n) return;

  const float bias = bp[j0 + lm] + bn[j0 + lm];
  v8f acc;
#pragma unroll
  for (int v = 0; v < 8; ++v) acc[v] = bias;

  const float* aprow = Ap + (size_t)(i0 + lm) * H_DIM + 2 * half;
  const float* wprow = Wp + (size_t)(j0 + lm) * H_DIM + 2 * half;
  const float* anrow = An + (size_t)(i0 + lm) * H_DIM + 2 * half;
  const float* wnrow = Wn + (size_t)(j0 + lm) * H_DIM + 2 * half;

#pragma unroll 4
  for (int k0 = 0; k0 < H_DIM; k0 += 4) {
    v2f ap = *(const v2f*)(aprow + k0);
    v2f wp = *(const v2f*)(wprow + k0);
    v2f an = *(const v2f*)(anrow + k0);
    v2f wn = *(const v2f*)(wnrow + k0);
    acc = __builtin_amdgcn_wmma_f32_16x16x4_f32(false, ap, false, wp,
                                                (short)0, acc, false, false);
    acc = __builtin_amdgcn_wmma_f32_16x16x4_f32(false, an, false, wn,
                                                (short)0, acc, false, false);
  }

#pragma unroll
  for (int v = 0; v < 8; ++v)
    out[(size_t)(i0 + v + 8 * half) * H_DIM + j0 + lm] = tanhf(acc[v]);
}

// ---------------------------------------------------------------------------
// SpMM scatter: out[row[e]] += val[e] * h[col[e]]
// One wave32 per edge; each lane owns a float4 (32 lanes * 4 = 128 channels).
// Gather is GLOBAL_LOAD_B128 from h (51.2 MB -> resident in 192 MB L2);
// scatter via hardware GLOBAL_ATOMIC_ADD_F32 (unsafeAtomicAdd).
// ---------------------------------------------------------------------------
__global__ void __launch_bounds__(256)
spmm_kernel(const int* __restrict__ row, const int* __restrict__ col,
            const float* __restrict__ val, const float* __restrict__ h,
            float* __restrict__ out, int e) {
  const int edge = blockIdx.x * (blockDim.x >> 5) + (threadIdx.x >> 5);
  const int lane = threadIdx.x & 31;
  if (edge >= e) return;

  const int   c = col[edge];
  const int   r = row[edge];
  const float v = val[edge];

  const float4* src = (const float4*)(h + (size_t)c * H_DIM);
  float4 m = src[lane];

  float* dst = out + (size_t)r * H_DIM + lane * 4;
  unsafeAtomicAdd(dst + 0, v * m.x);
  unsafeAtomicAdd(dst + 1, v * m.y);
  unsafeAtomicAdd(dst + 2, v * m.z);
  unsafeAtomicAdd(dst + 3, v * m.w);
}

// ---------------------------------------------------------------------------
// Orchestration. Inputs (setup_inputs order):
//  0:x  1:pos_row 2:pos_col 3:pos_val  4:neg_row 5:neg_col 6:neg_val
//  7:W_in 8:b_in  9:W_pos[L,H,H] 10:b_pos[L,H] 11:W_neg 12:b_neg
// h lives in d_out (updated in place each layer); hp/hn live in d_ws.
// ---------------------------------------------------------------------------
extern "C" void kernel_launch(void* const* d_in, const int* in_sizes, int n_in,
                              void* d_out, int out_size, void* d_ws, size_t ws_size,
                              hipStream_t stream) {
  const float* x       = (const float*)d_in[0];
  const int*   pos_row = (const int*)d_in[1];
  const int*   pos_col = (const int*)d_in[2];
  const float* pos_val = (const float*)d_in[3];
  const int*   neg_row = (const int*)d_in[4];
  const int*   neg_col = (const int*)d_in[5];
  const float* neg_val = (const float*)d_in[6];
  const float* W_in    = (const float*)d_in[7];
  const float* b_in    = (const float*)d_in[8];
  const float* W_pos   = (const float*)d_in[9];
  const float* b_pos   = (const float*)d_in[10];
  const float* W_neg   = (const float*)d_in[11];
  const float* b_neg   = (const float*)d_in[12];

  float* h  = (float*)d_out;                       // [N, H]
  float* hp = (float*)d_ws;                        // [N, H]
  float* hn = hp + (size_t)N_NODES * H_DIM;        // [N, H]

  const dim3 blk(256);
  const int row_tiles   = (N_NODES + 15) / 16;     // 6250 (exact)
  const int spmm_blocks = (E_EDGES + 7) / 8;       // 8 edges (waves) per block

  // h = tanh(x @ W_in^T + b_in)
  gemm_tanh_kernel<<<row_tiles, blk, 0, stream>>>(x, W_in, b_in, h, N_NODES);

  for (int l = 0; l < L_LAYERS; ++l) {
    // zero hp and hn (contiguous) before atomic accumulation
    hipMemsetAsync(hp, 0, 2ull * N_NODES * H_DIM * sizeof(float), stream);

    spmm_kernel<<<spmm_blocks, blk, 0, stream>>>(pos_row, pos_col, pos_val,
                                                 h, hp, E_EDGES);
    spmm_kernel<<<spmm_blocks, blk, 0, stream>>>(neg_row, neg_col, neg_val,
                                                 h, hn, E_EDGES);

    gemm2_tanh_kernel<<<row_tiles, blk, 0, stream>>>(
        hp, W_pos + (size_t)l * H_DIM * H_DIM, b_pos + (size_t)l * H_DIM,
        hn, W_neg + (size_t)l * H_DIM * H_DIM, b_neg + (size_t)l * H_DIM,
        h, N_NODES);
  }
}